// CollisionAvoidanceGNN_19250043420762
// MI455X (gfx1250) — compile-verified
//
#include <hip/hip_runtime.h>
#include <stdint.h>

// ---------------------------------------------------------------------------
// 2-layer GCN (PyG GCNConv semantics) for MI455X / gfx1250.
//   layer1: h1 = x@W1 ; agg1 = A_hat @ h1 + b1 ; (ReLU deferred to gemm2 load)
//   layer2: h2 = relu(agg1)@W2 ; out = A_hat @ h2 + b2
// A_hat = D^-1/2 (A + I) D^-1/2 applied as per-edge gather/scale/atomic-scatter.
// GEMMs use V_WMMA_F32_16X16X4_F32 (full f32 precision).
// All padding-with-zero is done via value selects on unconditional in-bounds
// loads (v_cndmask), so no EXEC-mask branches appear around WMMA operands.
// ---------------------------------------------------------------------------

typedef float v2f __attribute__((ext_vector_type(2)));
typedef float v8f __attribute__((ext_vector_type(8)));

#define TPB 256

// ---- degree / normalization -----------------------------------------------

__global__ void k_deg_init(float* __restrict__ deg, int n) {
  int i = blockIdx.x * blockDim.x + threadIdx.x;
  if (i < n) deg[i] = 1.0f;  // self-loop contribution
}

__global__ void k_deg_count(const long long* __restrict__ dst,
                            float* __restrict__ deg, int e) {
  int i = blockIdx.x * blockDim.x + threadIdx.x;
  if (i < e) {
    __builtin_prefetch(&dst[i + 4096], 0, 1);
    atomicAdd(&deg[(int)dst[i]], 1.0f);
  }
}

__global__ void k_dinv(float* __restrict__ deg, int n) {
  int i = blockIdx.x * blockDim.x + threadIdx.x;
  if (i < n) deg[i] = rsqrtf(deg[i]);  // in-place: deg -> d^-1/2
}

// ---- layer-1 GEMM: h1[n,32] = x[n,3] @ W1[3,32] via WMMA f32 16x16x4 ------
// One wave handles 16 nodes; two 16-column tiles -> 2 WMMAs.

__global__ void k_gemm1(const float* __restrict__ x, const float* __restrict__ W1,
                        float* __restrict__ h1, int n) {
  int wave = (blockIdx.x * blockDim.x + threadIdx.x) >> 5;
  int lane = threadIdx.x & 31;
  int base = wave * 16;
  if (base >= n) return;            // wave-uniform guard (n % 16 == 0)
  int h = lane >> 4;                // lane half: K-pair selector
  int m = lane & 15;                // row (A) / column (B,C,D) index

  // A tile 16x4 (K=3 padded with 0): lanes 0-15 hold K0,K1; lanes 16-31 K2,K3.
  // Unconditional in-bounds loads; zero-pad via value select (no exec branch).
  int row = base + m;
  float ax = x[row * 3 + 2 * h];    // h=0 -> K0, h=1 -> K2
  float ay = x[row * 3 + 1];        // K1 (only meaningful for h=0)
  v2f a;
  a.x = ax;
  a.y = h ? 0.0f : ay;              // K3 = 0 padding for upper half

  #pragma unroll
  for (int t = 0; t < 2; ++t) {     // two tiles of 16 output columns
    int col = t * 16 + m;
    // B tile 4x16: VGPR j holds K=j (lanes 0-15) / K=j+2 (lanes 16-31)
    float bx = W1[2 * h * 32 + col];  // h=0 -> W1 row0, h=1 -> row2
    float by = W1[1 * 32 + col];      // W1 row1
    v2f b;
    b.x = bx;
    b.y = h ? 0.0f : by;              // K3 = 0 padding
    v8f c = {};
    c = __builtin_amdgcn_wmma_f32_16x16x4_f32(false, a, false, b,
                                              (short)0, c, false, false);
    // D 16x16: VGPR r -> row (r + 8*h), col m
    #pragma unroll
    for (int r = 0; r < 8; ++r)
      h1[(size_t)(base + r + 8 * h) * 32 + col] = c[r];
  }
}

// ---- layer-1 aggregation ---------------------------------------------------

__global__ void k_agg1_init(const float* __restrict__ h1,
                            const float* __restrict__ dinv,
                            const float* __restrict__ b1,
                            float* __restrict__ agg1, int n32) {
  int i = blockIdx.x * blockDim.x + threadIdx.x;
  if (i < n32) {
    int node = i >> 5, ch = i & 31;
    float di = dinv[node];
    agg1[i] = h1[i] * di * di + b1[ch];  // self-loop + bias
  }
}

__global__ void k_agg1_edges(const long long* __restrict__ src,
                             const long long* __restrict__ dst,
                             const float* __restrict__ h1,
                             const float* __restrict__ dinv,
                             float* __restrict__ agg1, int e) {
  int i = blockIdx.x * blockDim.x + threadIdx.x;
  if (i >= e) return;
  __builtin_prefetch(&src[i + 4096], 0, 1);
  __builtin_prefetch(&dst[i + 4096], 0, 1);
  int s = (int)src[i], d = (int)dst[i];
  float norm = dinv[s] * dinv[d];
  const float4* hs = (const float4*)&h1[(size_t)s * 32];  // 128B row, L2-hot
  float* out = &agg1[(size_t)d * 32];
  #pragma unroll
  for (int q = 0; q < 8; ++q) {
    float4 v = hs[q];
    atomicAdd(&out[q * 4 + 0], v.x * norm);
    atomicAdd(&out[q * 4 + 1], v.y * norm);
    atomicAdd(&out[q * 4 + 2], v.z * norm);
    atomicAdd(&out[q * 4 + 3], v.w * norm);
  }
}

// ---- layer-2 GEMM: h2[n,3] = relu(agg1)[n,32] @ W2[32,3] -------------------
// One wave, 16 nodes, K=32 as 8 chained 16x16x4 WMMAs; store cols 0..2.

__global__ void k_gemm2(const float* __restrict__ agg1, const float* __restrict__ W2,
                        float* __restrict__ h2, int n) {
  int wave = (blockIdx.x * blockDim.x + threadIdx.x) >> 5;
  int lane = threadIdx.x & 31;
  int base = wave * 16;
  if (base >= n) return;            // wave-uniform guard
  int h = lane >> 4;
  int m = lane & 15;
  int row = base + m;
  bool valid_col = (m < 3);
  int mc = valid_col ? m : 0;       // clamped column -> all B loads in bounds

  v8f c = {};
  #pragma unroll
  for (int kk = 0; kk < 8; ++kk) {
    int k0 = kk * 4 + h * 2;        // this lane-half's K pair
    // A: ReLU fused into the load (bias was folded into agg1 already)
    float2 av = *(const float2*)&agg1[(size_t)row * 32 + k0];
    v2f a;
    a.x = fmaxf(av.x, 0.0f);
    a.y = fmaxf(av.y, 0.0f);
    // B: W2 is [32,3] row-major; columns 3..15 are zero padding.
    // Unconditional clamped loads + value select (no exec branch, no OOB).
    float bx = W2[k0 * 3 + mc];
    float by = W2[(k0 + 1) * 3 + mc];
    v2f b;
    b.x = valid_col ? bx : 0.0f;
    b.y = valid_col ? by : 0.0f;
    c = __builtin_amdgcn_wmma_f32_16x16x4_f32(false, a, false, b,
                                              (short)0, c, false, false);
  }
  if (valid_col) {
    #pragma unroll
    for (int r = 0; r < 8; ++r)
      h2[(size_t)(base + r + 8 * h) * 3 + m] = c[r];
  }
}

// ---- layer-2 aggregation (writes final output) -----------------------------

__global__ void k_out_init(const float* __restrict__ h2,
                           const float* __restrict__ dinv,
                           const float* __restrict__ b2,
                           float* __restrict__ out, int n3) {
  int i = blockIdx.x * blockDim.x + threadIdx.x;
  if (i < n3) {
    int node = i / 3, ch = i - node * 3;
    float di = dinv[node];
    out[i] = h2[i] * di * di + b2[ch];  // self-loop + bias
  }
}

__global__ void k_out_edges(const long long* __restrict__ src,
                            const long long* __restrict__ dst,
                            const float* __restrict__ h2,
                            const float* __restrict__ dinv,
                            float* __restrict__ out, int e) {
  int i = blockIdx.x * blockDim.x + threadIdx.x;
  if (i >= e) return;
  __builtin_prefetch(&src[i + 4096], 0, 1);
  __builtin_prefetch(&dst[i + 4096], 0, 1);
  int s = (int)src[i], d = (int)dst[i];
  float norm = dinv[s] * dinv[d];
  float v0 = h2[(size_t)s * 3 + 0];
  float v1 = h2[(size_t)s * 3 + 1];
  float v2 = h2[(size_t)s * 3 + 2];
  atomicAdd(&out[(size_t)d * 3 + 0], v0 * norm);
  atomicAdd(&out[(size_t)d * 3 + 1], v1 * norm);
  atomicAdd(&out[(size_t)d * 3 + 2], v2 * norm);
}

// ---------------------------------------------------------------------------

extern "C" void kernel_launch(void* const* d_in, const int* in_sizes, int n_in,
                              void* d_out, int out_size, void* d_ws, size_t ws_size,
                              hipStream_t stream) {
  const float*     x  = (const float*)d_in[0];
  const long long* ei = (const long long*)d_in[1];  // int64 [2, E]
  const float*     W1 = (const float*)d_in[2];
  const float*     b1 = (const float*)d_in[3];
  const float*     W2 = (const float*)d_in[4];
  const float*     b2 = (const float*)d_in[5];
  float* out = (float*)d_out;

  const int n = in_sizes[0] / 3;
  const int e = in_sizes[1] / 2;
  const long long* src = ei;
  const long long* dst = ei + e;

  // workspace layout (all 128B aligned): dinv | h1 | agg1 | h2
  char* ws = (char*)d_ws;
  size_t off = 0;
  float* dinv = (float*)(ws + off); off += ((size_t)n * 4 + 127) & ~(size_t)127;
  float* h1   = (float*)(ws + off); off += (size_t)n * 32 * 4;
  float* agg1 = (float*)(ws + off); off += (size_t)n * 32 * 4;
  float* h2   = (float*)(ws + off); off += (((size_t)n * 3 * 4 + 127) & ~(size_t)127);
  (void)ws_size;

  const int bn    = (n + TPB - 1) / TPB;
  const int be    = (e + TPB - 1) / TPB;
  const int waves = (n + 15) / 16;                 // 16 nodes per wave32
  const int bg    = (waves * 32 + TPB - 1) / TPB;
  const int n32   = n * 32;
  const int n3    = n * 3;
  const int bn32  = (n32 + TPB - 1) / TPB;
  const int bn3   = (n3 + TPB - 1) / TPB;

  // normalization
  k_deg_init <<<bn, TPB, 0, stream>>>(dinv, n);
  k_deg_count<<<be, TPB, 0, stream>>>(dst, dinv, e);
  k_dinv     <<<bn, TPB, 0, stream>>>(dinv, n);

  // layer 1
  k_gemm1     <<<bg,   TPB, 0, stream>>>(x, W1, h1, n);
  k_agg1_init <<<bn32, TPB, 0, stream>>>(h1, dinv, b1, agg1, n32);
  k_agg1_edges<<<be,   TPB, 0, stream>>>(src, dst, h1, dinv, agg1, e);

  // layer 2
  k_gemm2    <<<bg,  TPB, 0, stream>>>(agg1, W2, h2, n);
  k_out_init <<<bn3, TPB, 0, stream>>>(h2, dinv, b2, out, n3);
  k_out_edges<<<be,  TPB, 0, stream>>>(src, dst, h2, dinv, out, e);
}